// GNN_PRETRAIN_60584808677997
// MI455X (gfx1250) — compile-verified
//
#include <hip/hip_runtime.h>

// ---------------------------------------------------------------------------
// GraphSAGE 2-layer forward on MI455X (gfx1250).
//   agg = segment_sum(x[src], dst) * inv_deg
//   out = agg @ Wl^T + b + x @ Wr^T
// Scatter phase: L2-resident gather + global_atomic_add_f32 (dominant cost).
// GEMM phase: V_WMMA_F32_16X16X4_F32 (exact fp32, matches reference numerics).
// ---------------------------------------------------------------------------

typedef __attribute__((ext_vector_type(2))) float v2f;
typedef __attribute__((ext_vector_type(8))) float v8f;

#define D 64  // feature dim (both layers)

// ---------------- degree = segment_sum(ones, dst) ----------------
__global__ __launch_bounds__(256) void degree_kernel(
    const long long* __restrict__ dst, float* __restrict__ deg, int n_edges) {
  int e = blockIdx.x * blockDim.x + threadIdx.x;
  if (e < n_edges) {
    atomicAdd(&deg[(int)dst[e]], 1.0f);
  }
}

// ---------------- inv_deg = 1 / max(deg, 1)  (in place) ----------------
__global__ __launch_bounds__(256) void invdeg_kernel(
    float* __restrict__ deg, int n_nodes) {
  int i = blockIdx.x * blockDim.x + threadIdx.x;
  if (i < n_nodes) {
    float d = deg[i];
    deg[i] = 1.0f / fmaxf(d, 1.0f);
  }
}

// ---------------- scatter-add: agg[dst] += x[src], 16 threads / edge -------
// Each thread handles one float4 slice: b128 gather, 4 coalesced f32 atomics.
__global__ __launch_bounds__(256) void scatter_add_kernel(
    const float* __restrict__ x, const long long* __restrict__ src,
    const long long* __restrict__ dst, float* __restrict__ agg, int n_edges) {
  long long tid = (long long)blockIdx.x * blockDim.x + threadIdx.x;
  long long e = tid >> 4;          // edge
  int c4 = (int)(tid & 15);        // which float4 of 16
  if (e >= n_edges) return;
  int s = (int)src[e];
  int d = (int)dst[e];
  const float4 v = *(const float4*)(x + (size_t)s * D + c4 * 4);
  float* o = agg + (size_t)d * D + c4 * 4;
  atomicAdd(o + 0, v.x);
  atomicAdd(o + 1, v.y);
  atomicAdd(o + 2, v.z);
  atomicAdd(o + 3, v.w);
}

// ---------------- SAGE layer via WMMA f32 16x16x4 ----------------
// One wave32 per 16-node tile; 8 waves (256 thr) per block.
// out[16,64] = (agg*inv_deg)[16,64] @ Wl^T + b + x[16,64] @ Wr^T
// A frag (16x4 f32, 2 VGPRs): lane&15 = M row; lane>=16 -> K pair {k0+2,k0+3}.
// B frag (4x16 f32, 2 VGPRs): lane&15 = N col; lane>=16 -> K pair {k0+2,k0+3}.
// C/D (16x16 f32, 8 VGPRs):  acc[r] = C[r + 8*(lane>=16)][lane&15].
__global__ __launch_bounds__(256) void sage_wmma_kernel(
    const float* __restrict__ xin,      // [N,64] self features
    const float* __restrict__ agg,      // [N,64] neighbor sums
    const float* __restrict__ inv_deg,  // [N]
    const float* __restrict__ Wl,       // [64,64] row-major (out,in)
    const float* __restrict__ bias,     // [64]
    const float* __restrict__ Wr,       // [64,64]
    float* __restrict__ out,            // [N,64]
    int n_tiles) {
  const int lane = threadIdx.x & 31;
  const int wave = threadIdx.x >> 5;
  const int tile = blockIdx.x * 8 + wave;
  if (tile >= n_tiles) return;  // wave-uniform: EXEC all-1s below (WMMA req)

  const int row0 = tile * 16;
  const int half = lane >> 4;   // 0: K pair {0,1}; 1: K pair {2,3}
  const int l15 = lane & 15;
  const int arow = row0 + l15;
  const float id = inv_deg[arow];

  // Accumulators init with bias: C[m][n] = b[n] for all m.
  v8f acc[4];
#pragma unroll
  for (int nt = 0; nt < 4; ++nt) {
    float bv = bias[nt * 16 + l15];
#pragma unroll
    for (int r = 0; r < 8; ++r) acc[nt][r] = bv;
  }

  const float* aggrow = agg + (size_t)arow * D;
  const float* xrow = xin + (size_t)arow * D;

#pragma unroll
  for (int k0 = 0; k0 < D; k0 += 4) {
    const int kb = k0 + half * 2;
    v2f a_agg, a_x;
    a_agg.x = aggrow[kb] * id;
    a_agg.y = aggrow[kb + 1] * id;
    a_x.x = xrow[kb];
    a_x.y = xrow[kb + 1];
#pragma unroll
    for (int nt = 0; nt < 4; ++nt) {
      const int ncol = nt * 16 + l15;  // B[k][n] = W[n][k]
      const float* wl = Wl + (size_t)ncol * D + kb;
      const float* wr = Wr + (size_t)ncol * D + kb;
      v2f bl, br;
      bl.x = wl[0];
      bl.y = wl[1];
      br.x = wr[0];
      br.y = wr[1];
      acc[nt] = __builtin_amdgcn_wmma_f32_16x16x4_f32(
          false, a_agg, false, bl, (short)0, acc[nt], false, false);
      acc[nt] = __builtin_amdgcn_wmma_f32_16x16x4_f32(
          false, a_x, false, br, (short)0, acc[nt], false, false);
    }
  }

  // Store D tile: M = r + 8*half, N = nt*16 + l15.
#pragma unroll
  for (int nt = 0; nt < 4; ++nt) {
#pragma unroll
    for (int r = 0; r < 8; ++r) {
      const int m = r + 8 * half;
      out[(size_t)(row0 + m) * D + nt * 16 + l15] = acc[nt][r];
    }
  }
}

extern "C" void kernel_launch(void* const* d_in, const int* in_sizes, int n_in,
                              void* d_out, int out_size, void* d_ws,
                              size_t ws_size, hipStream_t stream) {
  (void)n_in;
  (void)out_size;
  (void)ws_size;
  const float* x = (const float*)d_in[0];
  const long long* ei = (const long long*)d_in[1];  // int64 [2, E]
  const float* Wl1 = (const float*)d_in[2];
  const float* b1 = (const float*)d_in[3];
  const float* Wr1 = (const float*)d_in[4];
  const float* Wl2 = (const float*)d_in[5];
  const float* b2 = (const float*)d_in[6];
  const float* Wr2 = (const float*)d_in[7];
  float* out = (float*)d_out;

  const int N = in_sizes[0] / D;  // 100000
  const int E = in_sizes[1] / 2;  // 1600000
  const long long* src = ei;
  const long long* dst = ei + E;

  // Workspace carve-up (needs ~53 MB).
  const size_t aggBytes = (size_t)N * D * sizeof(float);
  char* ws = (char*)d_ws;
  float* inv_deg = (float*)ws;                                // N floats
  float* agg = (float*)(ws + (1 << 20));                      // N*64 floats
  float* h = (float*)(ws + (1 << 20) + ((aggBytes + 4095) & ~(size_t)4095));

  const int n_tiles = N / 16;  // 100000 % 16 == 0
  dim3 blk(256);
  dim3 gDeg((E + 255) / 256);
  dim3 gNode((N + 255) / 256);
  dim3 gScat((unsigned)(((long long)E * 16 + 255) / 256));
  dim3 gSage((n_tiles + 7) / 8);

  // Degrees (shared by both layers).
  hipMemsetAsync(inv_deg, 0, (size_t)N * sizeof(float), stream);
  degree_kernel<<<gDeg, blk, 0, stream>>>(dst, inv_deg, E);
  invdeg_kernel<<<gNode, blk, 0, stream>>>(inv_deg, N);

  // Layer 1: x -> h
  hipMemsetAsync(agg, 0, aggBytes, stream);
  scatter_add_kernel<<<gScat, blk, 0, stream>>>(x, src, dst, agg, E);
  sage_wmma_kernel<<<gSage, blk, 0, stream>>>(x, agg, inv_deg, Wl1, b1, Wr1, h,
                                              n_tiles);

  // Layer 2: h -> out
  hipMemsetAsync(agg, 0, aggBytes, stream);
  scatter_add_kernel<<<gScat, blk, 0, stream>>>(h, src, dst, agg, E);
  sage_wmma_kernel<<<gSage, blk, 0, stream>>>(h, agg, inv_deg, Wl2, b2, Wr2,
                                              out, n_tiles);
}